// GATNet_ori_33930241638749
// MI455X (gfx1250) — compile-verified
//
#include <hip/hip_runtime.h>
#include <hip/hip_bf16.h>
#include <math.h>

// ---------------------------------------------------------------------------
// 2-layer GAT (PyG GATConv semantics) + global mean pool, for MI455X gfx1250.
//   L1: h1 = x@W1 [N,32] (8 heads x 4), edge softmax per head, ELU
//   L2: h2 = x2@W2 [N,64] (1 head), edge softmax, +b2, mean-pool over 64 graphs
// Dense projections use V_WMMA_F32_16X16X4_F32 (exact fp32 on the matrix pipe).
// Edge phase: wave-per-edge coalesced gathers + L2-resident float atomics.
// ---------------------------------------------------------------------------

typedef float v2f __attribute__((ext_vector_type(2)));
typedef float v8f __attribute__((ext_vector_type(8)));

#define NEG_SLOPE 0.2f

// ---------------- utility ----------------

__global__ void fill_kernel(float* __restrict__ p, float v, long n) {
  long i = (long)blockIdx.x * blockDim.x + threadIdx.x;
  long stride = (long)gridDim.x * blockDim.x;
  for (; i < n; i += stride) p[i] = v;
}

__device__ __forceinline__ float atomic_max_f32(float* addr, float value) {
  // Bit-trick float max: positive floats order as signed ints, negative floats
  // order inversely as unsigned ints. Init value must be -inf (0xFF800000).
  if (value >= 0.0f)
    return __int_as_float(atomicMax((int*)addr, __float_as_int(value)));
  else
    return __uint_as_float(atomicMin((unsigned int*)addr, __float_as_uint(value)));
}

__device__ __forceinline__ void edge_sd(const int* __restrict__ ei, int E, int e,
                                        int& s, int& d) {
  if (e < E) { s = ei[e]; d = ei[E + e]; }
  else       { s = e - E; d = e - E; }     // appended self loops
}

__device__ __forceinline__ float leaky(float x) {
  return x > 0.0f ? x : NEG_SLOPE * x;
}

// ---------------- WMMA fp32 GEMM: C[M,Nc] = A[M,K] @ B[K,Nc] ----------------
// One wave per 16x16 output tile; K stepped by 4 with V_WMMA_F32_16X16X4_F32.
// Requires M % 16 == 0, K % 4 == 0, Nc % 16 == 0 (true here: 100000,128/32,32/64).
// A 16x4 layout : lanes 0-15 -> K={kk,kk+1}, lanes 16-31 -> K={kk+2,kk+3}, M=lane&15
// B 4x16 layout : same K split per lane half, col = lane&15
// C 16x16 layout: vgpr r -> M = r + 8*(lane>>4), col = lane&15
__global__ void gemm_wmma_f32(const float* __restrict__ A,
                              const float* __restrict__ B,
                              float* __restrict__ C,
                              int M, int K, int Nc) {
  const int wave = threadIdx.x >> 5;
  const int lane = threadIdx.x & 31;
  const int tileM = blockIdx.x * (blockDim.x >> 5) + wave;
  const int numTilesM = M >> 4;
  if (tileM >= numTilesM) return;            // wave-uniform: EXEC stays all-ones
  const int n0 = blockIdx.y << 4;
  const int col = n0 + (lane & 15);
  const int rowA = (tileM << 4) + (lane & 15);
  const int khalf = (lane >> 4) << 1;        // 0 or 2

  v8f c = {};
  const float* __restrict__ ap = A + (long)rowA * K + khalf;
  for (int kk = 0; kk < K; kk += 4) {
    v2f a, b;
    a.x = ap[kk];
    a.y = ap[kk + 1];
    b.x = B[(long)(kk + khalf) * Nc + col];
    b.y = B[(long)(kk + khalf + 1) * Nc + col];
    c = __builtin_amdgcn_wmma_f32_16x16x4_f32(false, a, false, b,
                                              (short)0, c, false, false);
  }

  const int mbase = (tileM << 4) + ((lane >> 4) << 3);
#pragma unroll
  for (int r = 0; r < 8; ++r)
    C[(long)(mbase + r) * Nc + col] = c[r];
}

// ---------------- Layer 1 (8 heads x dim 4) ----------------

// a_s[n,h] = <h1[n,h,:], att_src[h,:]>, a_d likewise. One thread per (node,head).
__global__ void attn_coef1(const float* __restrict__ h1,
                           const float* __restrict__ att_s,
                           const float* __restrict__ att_d,
                           float* __restrict__ a_s, float* __restrict__ a_d,
                           int total /* N*8 */) {
  int t = blockIdx.x * blockDim.x + threadIdx.x;
  if (t >= total) return;
  int node = t >> 3, hd = t & 7;
  const float* hp = h1 + (long)node * 32 + hd * 4;
  const float* ws = att_s + hd * 4;
  const float* wd = att_d + hd * 4;
  float s = 0.f, d = 0.f;
#pragma unroll
  for (int f = 0; f < 4; ++f) { s += hp[f] * ws[f]; d += hp[f] * wd[f]; }
  a_s[t] = s;
  a_d[t] = d;
}

// Pass 1: per-destination softmax max (atomic). One thread per (edge,head).
__global__ void edge_max1(const int* __restrict__ ei, int E, int Et,
                          const float* __restrict__ a_s,
                          const float* __restrict__ a_d,
                          float* __restrict__ amax) {
  int t = blockIdx.x * blockDim.x + threadIdx.x;
  if (t >= Et * 8) return;
  int e = t >> 3, hd = t & 7, s, d;
  edge_sd(ei, E, e, s, d);
  float al = leaky(a_s[s * 8 + hd] + a_d[d * 8 + hd]);
  atomic_max_f32(&amax[d * 8 + hd], al);
}

// Pass 2 (fused): acc[dst,h,f] += ex * h1[src,h,f]; denom[dst,h] += ex.
// One wave per edge; lane covers feature column (hd = lane>>2).
__global__ void edge_acc1(const int* __restrict__ ei, int E, int Et,
                          const float* __restrict__ a_s,
                          const float* __restrict__ a_d,
                          const float* __restrict__ amax,
                          const float* __restrict__ h1,
                          float* __restrict__ acc, float* __restrict__ denom) {
  int gid = blockIdx.x * blockDim.x + threadIdx.x;
  int e = gid >> 5, lane = gid & 31;
  if (e >= Et) return;
  int hd = lane >> 2, s, d;
  edge_sd(ei, E, e, s, d);
  float al = leaky(a_s[s * 8 + hd] + a_d[d * 8 + hd]);
  float ex = expf(al - amax[d * 8 + hd]);
  atomicAdd(&acc[(long)d * 32 + lane], ex * h1[(long)s * 32 + lane]);
  if ((lane & 3) == 0) atomicAdd(&denom[d * 8 + hd], ex);
}

// Normalize, +bias, ELU (in place: acc -> x2). One thread per (node, col32).
__global__ void node1(float* __restrict__ acc, const float* __restrict__ denom,
                      const float* __restrict__ b1, int total /* N*32 */) {
  int t = blockIdx.x * blockDim.x + threadIdx.x;
  if (t >= total) return;
  int node = t >> 5, c = t & 31, hd = c >> 2;
  float v = acc[t] / (denom[node * 8 + hd] + 1e-16f) + b1[c];
  acc[t] = v > 0.0f ? v : (expf(v) - 1.0f);   // ELU(alpha=1)
}

// ---------------- Layer 2 (1 head x dim 64) ----------------

// One wave per node: 64-wide dot with att_src2/att_dst2 via shuffle reduction.
__global__ void attn_coef2(const float* __restrict__ h2,
                           const float* __restrict__ att_s,
                           const float* __restrict__ att_d,
                           float* __restrict__ a_s, float* __restrict__ a_d,
                           int Nn) {
  int gid = blockIdx.x * blockDim.x + threadIdx.x;
  int node = gid >> 5, lane = gid & 31;
  if (node >= Nn) return;
  const float* hp = h2 + (long)node * 64;
  float s = hp[lane] * att_s[lane] + hp[lane + 32] * att_s[lane + 32];
  float d = hp[lane] * att_d[lane] + hp[lane + 32] * att_d[lane + 32];
#pragma unroll
  for (int off = 16; off > 0; off >>= 1) {
    s += __shfl_down(s, off, 32);
    d += __shfl_down(d, off, 32);
  }
  if (lane == 0) { a_s[node] = s; a_d[node] = d; }
}

__global__ void edge_max2(const int* __restrict__ ei, int E, int Et,
                          const float* __restrict__ a_s,
                          const float* __restrict__ a_d,
                          float* __restrict__ amax) {
  int e = blockIdx.x * blockDim.x + threadIdx.x;
  if (e >= Et) return;
  int s, d;
  edge_sd(ei, E, e, s, d);
  float al = leaky(a_s[s] + a_d[d]);
  atomic_max_f32(&amax[d], al);
}

// One wave per edge; each lane handles features {lane, lane+32}.
__global__ void edge_acc2(const int* __restrict__ ei, int E, int Et,
                          const float* __restrict__ a_s,
                          const float* __restrict__ a_d,
                          const float* __restrict__ amax,
                          const float* __restrict__ h2,
                          float* __restrict__ out2, float* __restrict__ denom) {
  int gid = blockIdx.x * blockDim.x + threadIdx.x;
  int e = gid >> 5, lane = gid & 31;
  if (e >= Et) return;
  int s, d;
  edge_sd(ei, E, e, s, d);
  float al = leaky(a_s[s] + a_d[d]);
  float ex = expf(al - amax[d]);
  const float* hp = h2 + (long)s * 64;
  float* op = out2 + (long)d * 64;
  atomicAdd(&op[lane], ex * hp[lane]);
  atomicAdd(&op[lane + 32], ex * hp[lane + 32]);
  if (lane == 0) atomicAdd(&denom[d], ex);
}

// Normalize + b2, then scatter into per-graph sums. One thread per (node,feat).
__global__ void final_pool_sum(const float* __restrict__ out2,
                               const float* __restrict__ denom,
                               const float* __restrict__ b2,
                               const int* __restrict__ batch,
                               float* __restrict__ gsum, int total /* N*64 */) {
  int t = blockIdx.x * blockDim.x + threadIdx.x;
  if (t >= total) return;
  int node = t >> 6, f = t & 63;
  float v = out2[t] / (denom[node] + 1e-16f) + b2[f];
  atomicAdd(&gsum[batch[node] * 64 + f], v);
}

__global__ void count_kernel(const int* __restrict__ batch, float* __restrict__ cnt,
                             int Nn) {
  int n = blockIdx.x * blockDim.x + threadIdx.x;
  if (n >= Nn) return;
  atomicAdd(&cnt[batch[n]], 1.0f);
}

__global__ void pool_div(float* __restrict__ gsum, const float* __restrict__ cnt,
                         int total /* G*64 */) {
  int t = blockIdx.x * blockDim.x + threadIdx.x;
  if (t >= total) return;
  float c = cnt[t >> 6];
  gsum[t] /= (c > 1.0f ? c : 1.0f);
}

// ---------------- launch ----------------

extern "C" void kernel_launch(void* const* d_in, const int* in_sizes, int n_in,
                              void* d_out, int out_size, void* d_ws, size_t ws_size,
                              hipStream_t stream) {
  const float* x   = (const float*)d_in[0];
  const int*   ei  = (const int*)  d_in[1];
  const int*   bat = (const int*)  d_in[2];
  const float* W1  = (const float*)d_in[3];
  const float* as1 = (const float*)d_in[4];
  const float* ad1 = (const float*)d_in[5];
  const float* b1  = (const float*)d_in[6];
  const float* W2  = (const float*)d_in[7];
  const float* as2 = (const float*)d_in[8];
  const float* ad2 = (const float*)d_in[9];
  const float* b2  = (const float*)d_in[10];
  float* out = (float*)d_out;

  const int Nn = in_sizes[0] / 128;   // 100000 (divisible by 16)
  const int E  = in_sizes[1] / 2;     // 3200000
  const int Et = E + Nn;              // edges + self loops
  const int G64 = out_size;           // G * 64

  // Workspace layout (floats), total = 228*N + G
  const long N = (long)Nn;
  float* ws  = (float*)d_ws;
  float* h1  = ws;              // 32N
  float* aS1 = ws + 32 * N;     // 8N
  float* aD1 = ws + 40 * N;     // 8N
  float* mx1 = ws + 48 * N;     // 8N
  float* dn1 = ws + 56 * N;     // 8N
  float* x2  = ws + 64 * N;     // 32N (accum -> activated)
  float* h2  = ws + 96 * N;     // 64N
  float* aS2 = ws + 160 * N;    // N
  float* aD2 = ws + 161 * N;    // N
  float* mx2 = ws + 162 * N;    // N
  float* dn2 = ws + 163 * N;    // N
  float* o2  = ws + 164 * N;    // 64N
  float* cnt = ws + 228 * N;    // G

  const int B = 256;
  const int FB = 2048;  // fill grid

  // ---- init accumulators (every call; deterministic, graph-safe) ----
  fill_kernel<<<FB, B, 0, stream>>>(mx1, -INFINITY, 8 * N);
  fill_kernel<<<FB, B, 0, stream>>>(dn1, 0.0f, 40 * N);          // dn1 + x2
  fill_kernel<<<FB, B, 0, stream>>>(mx2, -INFINITY, N);
  fill_kernel<<<FB, B, 0, stream>>>(dn2, 0.0f, 65 * N + (G64 >> 6)); // dn2+o2+cnt
  fill_kernel<<<FB, B, 0, stream>>>(out, 0.0f, (long)G64);

  const int tilesM = Nn >> 4;                       // 6250
  const int wavesPerBlk = B >> 5;                   // 8

  // ---- Layer 1 ----
  dim3 g1((tilesM + wavesPerBlk - 1) / wavesPerBlk, 32 / 16);
  gemm_wmma_f32<<<g1, B, 0, stream>>>(x, W1, h1, Nn, 128, 32);

  attn_coef1<<<(Nn * 8 + B - 1) / B, B, 0, stream>>>(h1, as1, ad1, aS1, aD1, Nn * 8);

  edge_max1<<<((long)Et * 8 + B - 1) / B, B, 0, stream>>>(ei, E, Et, aS1, aD1, mx1);

  edge_acc1<<<((long)Et * 32 + B - 1) / B, B, 0, stream>>>(ei, E, Et, aS1, aD1,
                                                           mx1, h1, x2, dn1);

  node1<<<(Nn * 32 + B - 1) / B, B, 0, stream>>>(x2, dn1, b1, Nn * 32);

  // ---- Layer 2 ----
  dim3 g2((tilesM + wavesPerBlk - 1) / wavesPerBlk, 64 / 16);
  gemm_wmma_f32<<<g2, B, 0, stream>>>(x2, W2, h2, Nn, 32, 64);

  attn_coef2<<<((long)Nn * 32 + B - 1) / B, B, 0, stream>>>(h2, as2, ad2, aS2, aD2, Nn);

  edge_max2<<<(Et + B - 1) / B, B, 0, stream>>>(ei, E, Et, aS2, aD2, mx2);

  edge_acc2<<<((long)Et * 32 + B - 1) / B, B, 0, stream>>>(ei, E, Et, aS2, aD2,
                                                           mx2, h2, o2, dn2);

  // ---- normalize + global mean pool ----
  final_pool_sum<<<(Nn * 64 + B - 1) / B, B, 0, stream>>>(o2, dn2, b2, bat, out,
                                                          Nn * 64);
  count_kernel<<<(Nn + B - 1) / B, B, 0, stream>>>(bat, cnt, Nn);
  pool_div<<<(G64 + B - 1) / B, B, 0, stream>>>(out, cnt, G64);
}